// GNN_1477468750555
// MI455X (gfx1250) — compile-verified
//
#include <hip/hip_runtime.h>
#include <cstddef>

typedef __attribute__((ext_vector_type(2))) float v2f;
typedef __attribute__((ext_vector_type(8))) float v8f;

#define NNODES 50000
#define NEDGES 800000

// ---------------- graph preprocessing ----------------
__global__ void k_deg(const float* __restrict__ ea, const int* __restrict__ dst,
                      float* __restrict__ deg, int e_count) {
  int e = blockIdx.x * blockDim.x + threadIdx.x;
  if (e < e_count) atomicAdd(&deg[dst[e]], ea[e]);
}

__global__ void k_dinv(float* __restrict__ deg, int n) {
  int i = blockIdx.x * blockDim.x + threadIdx.x;
  if (i < n) {
    float v = deg[i];
    deg[i] = v > 0.f ? rsqrtf(v) : 0.f;
  }
}

__global__ void k_norm(const float* __restrict__ dinv, const float* __restrict__ ea,
                       const int* __restrict__ src, const int* __restrict__ dst,
                       float* __restrict__ nrm, int e_count) {
  int e = blockIdx.x * blockDim.x + threadIdx.x;
  if (e < e_count) nrm[e] = dinv[src[e]] * ea[e] * dinv[dst[e]];
}

__global__ void k_concat(const float* __restrict__ x, const float* __restrict__ pos,
                         float* __restrict__ h, int total) {
  int t = blockIdx.x * blockDim.x + threadIdx.x;
  if (t >= total) return;
  int n = t >> 3, f = t & 7;
  h[t] = (f < 6) ? x[n * 6 + f] : pos[n * 2 + (f - 6)];
}

// ---------------- propagation: h_out[dst] += norm * h_in[src] ----------------
// F = 1<<LOGF features/node, 1<<LOGF threads/edge: contiguous row gathers,
// float atomics resolve in L2 (feature matrix 12.8MB << 192MB L2).
// Pure 32-bit indexing (max offset 3.2M elements).
template <int LOGF>
__global__ void k_prop(const float* __restrict__ hin, float* __restrict__ hout,
                       const int* __restrict__ src, const int* __restrict__ dst,
                       const float* __restrict__ nrm, int total) {
  int t = blockIdx.x * blockDim.x + threadIdx.x;
  if (t >= total) return;
  int e = t >> LOGF;
  int f = t & ((1 << LOGF) - 1);
  atomicAdd(&hout[(dst[e] << LOGF) + f], nrm[e] * hin[(src[e] << LOGF) + f]);
}

// ---------------- O[n, 0:OS] = bias (cols >= fv zero-padded) ----------------
__global__ void k_biasinit(float* __restrict__ O, const float* __restrict__ b,
                           int fv, int os, int total) {
  int t = blockIdx.x * blockDim.x + threadIdx.x;
  if (t >= total) return;
  int j = t % os;
  O[t] = (j < fv) ? b[j] : 0.f;
}

// ---------------- fp32 WMMA GEMM accumulate: O += H @ W ----------------
// One wave32 owns a 16-row x (TN*16)-col strip: TN v8f accumulators, one
// float2 A-fragment load feeds TN back-to-back V_WMMA_F32_16X16X4_F32 per
// K-step. GUARD=false -> unconditional, immediate-offset W loads (fv == OS).
// H: N x FIN row-major. W: FIN x fv row-major. O: N x OS (OS = TN*16 >= fv).
template <int FIN, int TN, bool GUARD>
__global__ void __launch_bounds__(256) k_mm(const float* __restrict__ H,
                                            const float* __restrict__ W,
                                            float* __restrict__ O,
                                            int tilesM, int fv) {
  constexpr int OS = TN * 16;
  int wave = blockIdx.x * (blockDim.x >> 5) + (threadIdx.x >> 5);
  if (wave >= tilesM) return;  // wave-uniform; EXEC all-1s around WMMA
  int lane = threadIdx.x & 31;
  int hi = lane >> 4;  // half-wave select
  int lo = lane & 15;
  int m0 = wave << 4;
  int rbase = (m0 + hi * 8) * OS + lo;  // C layout: VGPR r -> M=r / M=r+8

  v8f c[TN];
#pragma unroll
  for (int t = 0; t < TN; ++t)
#pragma unroll
    for (int r = 0; r < 8; ++r)
      c[t][r] = O[rbase + r * OS + t * 16];

  // A fragment: lanes 0-15 -> K = k0,k0+1 ; lanes 16-31 -> K = k0+2,k0+3
  const float* hrow = H + (m0 + lo) * FIN + hi * 2;
#pragma unroll
  for (int k0 = 0; k0 < FIN; k0 += 4) {
    v2f a = *(const v2f*)(hrow + k0);  // 8B-aligned float2
    int kr = k0 + hi * 2;
#pragma unroll
    for (int t = 0; t < TN; ++t) {
      int col = t * 16 + lo;
      v2f b;
      if (GUARD) {  // only layer 3 (fv=10 padded to 16)
        b.x = (col < fv) ? W[kr * fv + col] : 0.f;
        b.y = (col < fv) ? W[(kr + 1) * fv + col] : 0.f;
      } else {      // fv == OS: straight-line loads
        b.x = W[kr * OS + col];
        b.y = W[(kr + 1) * OS + col];
      }
      // (neg_a, A, neg_b, B, c_mod, C, reuse_a, reuse_b)
      c[t] = __builtin_amdgcn_wmma_f32_16x16x4_f32(false, a, false, b, (short)0,
                                                   c[t], false, false);
    }
  }

#pragma unroll
  for (int t = 0; t < TN; ++t)
#pragma unroll
    for (int r = 0; r < 8; ++r)
      O[rbase + r * OS + t * 16] = c[t][r];
}

// ---------------- epilogues ----------------
__global__ void k_relu(float* __restrict__ O, int total) {
  int t = blockIdx.x * blockDim.x + threadIdx.x;
  if (t < total) O[t] = fmaxf(O[t], 0.f);
}

__global__ void k_lsm(const float* __restrict__ O, float* __restrict__ out, int n) {
  int i = blockIdx.x * blockDim.x + threadIdx.x;
  if (i >= n) return;
  const float* r = O + i * 16;  // padded stride, 10 valid cols
  float m = r[0];
#pragma unroll
  for (int j = 1; j < 10; ++j) m = fmaxf(m, r[j]);
  float s = 0.f;
#pragma unroll
  for (int j = 0; j < 10; ++j) s += __expf(r[j] - m);
  float ls = __logf(s);
#pragma unroll
  for (int j = 0; j < 10; ++j) out[i * 10 + j] = r[j] - m - ls;
}

// ---------------- one TAGConv layer ----------------
// O = bias; O += H0 @ W[0]; for k=1..3: Hk = prop(Hk-1); O += Hk @ W[k]; relu?
template <int FIN, int TN, bool GUARD>
static void run_layer(float* in, float* scratch, float* Obuf,
                      const float* Wl, const float* bl, int fv, bool do_relu,
                      const int* src, const int* dst, const float* nrm,
                      hipStream_t stream) {
  constexpr int OS = TN * 16;
  constexpr int LOGF = (FIN == 8) ? 3 : 6;
  const int N = NNODES, E = NEDGES;
  int totO = N * OS;
  k_biasinit<<<(totO + 255) / 256, 256, 0, stream>>>(Obuf, bl, fv, OS, totO);
  int tilesM = N / 16;              // 3125 exactly
  int mmBlocks = (tilesM + 7) / 8;  // 8 waves per 256-thread block
  k_mm<FIN, TN, GUARD><<<mmBlocks, 256, 0, stream>>>(in, Wl, Obuf, tilesM, fv);
  float* cur = in;
  float* alt = scratch;
  int ptot = E * FIN;  // <= 51.2M, fits int
  int pBlocks = (ptot + 255) / 256;
  for (int k = 1; k <= 3; ++k) {
    hipMemsetAsync(alt, 0, (size_t)N * FIN * 4, stream);
    k_prop<LOGF><<<pBlocks, 256, 0, stream>>>(cur, alt, src, dst, nrm, ptot);
    float* tmp = cur; cur = alt; alt = tmp;  // h_{k-1} buffer becomes free
    k_mm<FIN, TN, GUARD><<<mmBlocks, 256, 0, stream>>>(
        cur, Wl + k * FIN * fv, Obuf, tilesM, fv);
  }
  if (do_relu) k_relu<<<(totO + 255) / 256, 256, 0, stream>>>(Obuf, totO);
}

extern "C" void kernel_launch(void* const* d_in, const int* in_sizes, int n_in,
                              void* d_out, int out_size, void* d_ws, size_t ws_size,
                              hipStream_t stream) {
  (void)in_sizes; (void)n_in; (void)out_size; (void)ws_size;
  const float* x   = (const float*)d_in[0];
  const float* pos = (const float*)d_in[1];
  const float* ea  = (const float*)d_in[2];
  const float* W1  = (const float*)d_in[3];
  const float* b1  = (const float*)d_in[4];
  const float* W2  = (const float*)d_in[5];
  const float* b2  = (const float*)d_in[6];
  const float* W3  = (const float*)d_in[7];
  const float* b3  = (const float*)d_in[8];
  const int*   ei  = (const int*)d_in[9];
  float* out = (float*)d_out;

  const int N = NNODES, E = NEDGES;
  const int* src = ei;      // edge_index[0]
  const int* dst = ei + E;  // edge_index[1]

  auto au = [](size_t v) { return (v + 255) & ~(size_t)255; };
  char* p = (char*)d_ws;
  float* deg  = (float*)p; p += au((size_t)N * 4);       // deg -> dinv (in place)
  float* nrm  = (float*)p; p += au((size_t)E * 4);       // edge norms
  float* bufA = (float*)p; p += au((size_t)N * 64 * 4);  // rotating feature bufs
  float* bufB = (float*)p; p += au((size_t)N * 64 * 4);
  float* bufO = (float*)p; p += au((size_t)N * 64 * 4);

  // --- normalization coefficients ---
  hipMemsetAsync(deg, 0, (size_t)N * 4, stream);
  k_deg<<<(E + 255) / 256, 256, 0, stream>>>(ea, dst, deg, E);
  k_dinv<<<(N + 255) / 256, 256, 0, stream>>>(deg, N);
  k_norm<<<(E + 255) / 256, 256, 0, stream>>>(deg, ea, src, dst, nrm, E);

  // --- h0 = concat(x, pos), width 8 ---
  k_concat<<<(N * 8 + 255) / 256, 256, 0, stream>>>(x, pos, bufA, N * 8);

  // layer1: 8 -> 64, relu    (in bufA, out bufO)
  run_layer<8, 4, false>(bufA, bufB, bufO, W1, b1, 64, true, src, dst, nrm, stream);
  // layer2: 64 -> 64, relu   (in bufO, out bufA)
  run_layer<64, 4, false>(bufO, bufB, bufA, W2, b2, 64, true, src, dst, nrm, stream);
  // layer3: 64 -> 10 (padded to 16), no relu (in bufA, out bufO)
  run_layer<64, 1, true>(bufA, bufB, bufO, W3, b3, 10, false, src, dst, nrm, stream);

  // log_softmax over 10 classes
  k_lsm<<<(N + 255) / 256, 256, 0, stream>>>(bufO, out, N);
}